// GLQuasiAlpha_Pooled_Enhanced_52055003627864
// MI455X (gfx1250) — compile-verified
//
#include <hip/hip_runtime.h>
#include <hip/hip_bf16.h>

// ---------------------------------------------------------------------------
// Problem constants
// ---------------------------------------------------------------------------
#define B_   32
#define CIN  16
#define COUT 32
#define H_   64
#define F_   39
#define D_   256

__device__ __forceinline__ int layerN(int l) { return (l == 0) ? 128 : 4096; }
__device__ __forceinline__ int wf_off(int l) { return (l == 0) ? 0 : ((l == 1) ? 131072 : 4325376); }

// workspace layout (floats)
#define OFF_BF  8519680   // 3*32*32*64
#define OFF_WST 8716288   // 3*1024*24
#define OFF_BLK 8790016   // 3*1024*64
#define OFF_BST 8986624   // 3*1024*18
#define OFF_GST 9041920   // 3*32*4

#define FLT_BIG 3.402823466e+38f

// ---------------------------------------------------------------------------
// CDNA5 WMMA f32 16x16x4 wrapper.
//  A 16x4 f32 : 2 VGPR/lane (lane<16 : row=lane, k={k0,k0+1}; lane>=16 : k={k0+2,k0+3})
//  B 4x16 f32 : 2 VGPR/lane (lane%16 = column, same k split as A)
//  C/D 16x16  : 8 VGPR/lane, VGPR j -> row j + 8*(lane>>4), col lane%16
// ---------------------------------------------------------------------------
typedef __attribute__((ext_vector_type(2))) float v2f;
typedef __attribute__((ext_vector_type(8))) float v8f;

__device__ __forceinline__ v8f wmma_f32_4(v2f a, v2f b, v8f c) {
  return __builtin_amdgcn_wmma_f32_16x16x4_f32(false, a, false, b, (short)0, c, false, false);
}

// ---------------------------------------------------------------------------
// helpers
// ---------------------------------------------------------------------------
__device__ __forceinline__ unsigned fkey(float x) {
  unsigned u = __float_as_uint(x);
  return (u & 0x80000000u) ? ~u : (u | 0x80000000u);
}
__device__ __forceinline__ float funkey(unsigned k) {
  return __uint_as_float((k & 0x80000000u) ? (k & 0x7FFFFFFFu) : ~k);
}

__device__ __forceinline__ float blockReduceSum(float v, float* red, int tid, int nth) {
  __syncthreads();
  red[tid] = v; __syncthreads();
  for (int s = nth >> 1; s > 0; s >>= 1) { if (tid < s) red[tid] += red[tid + s]; __syncthreads(); }
  return red[0];
}
__device__ __forceinline__ float blockReduceMin(float v, float* red, int tid, int nth) {
  __syncthreads();
  red[tid] = v; __syncthreads();
  for (int s = nth >> 1; s > 0; s >>= 1) { if (tid < s) red[tid] = fminf(red[tid], red[tid + s]); __syncthreads(); }
  return red[0];
}
__device__ __forceinline__ float blockReduceMax(float v, float* red, int tid, int nth) {
  __syncthreads();
  red[tid] = v; __syncthreads();
  for (int s = nth >> 1; s > 0; s >>= 1) { if (tid < s) red[tid] = fmaxf(red[tid], red[tid + s]); __syncthreads(); }
  return red[0];
}
__device__ __forceinline__ int blockReduceSumI(int v, int* red, int tid, int nth) {
  __syncthreads();
  red[tid] = v; __syncthreads();
  for (int s = nth >> 1; s > 0; s >>= 1) { if (tid < s) red[tid] += red[tid + s]; __syncthreads(); }
  return red[0];
}

// ascending bitonic sort of power-of-2 array in LDS
__device__ __forceinline__ void bitonic_sort(float* buf, int N, int tid, int nth) {
  for (int k = 2; k <= N; k <<= 1) {
    for (int j = k >> 1; j > 0; j >>= 1) {
      for (int i = tid; i < N; i += nth) {
        int ixj = i ^ j;
        if (ixj > i) {
          float x = buf[i], y = buf[ixj];
          bool up = ((i & k) == 0);
          if (up ? (x > y) : (x < y)) { buf[i] = y; buf[ixj] = x; }
        }
      }
      __syncthreads();
    }
  }
}

// ---------------------------------------------------------------------------
// K1: projection  wf = mapW @ w_flat + mapb ; bf = mapW @ b_flat + mapb
// grid 96 (l*32+b), block 128 (4 waves)
// ---------------------------------------------------------------------------
__global__ __launch_bounds__(128) void k_proj(
    const float* __restrict__ w0, const float* __restrict__ b0,
    const float* __restrict__ w1, const float* __restrict__ b1,
    const float* __restrict__ w2, const float* __restrict__ b2,
    const float* __restrict__ mapW, const float* __restrict__ mapb,
    float* __restrict__ ws) {
  int l = blockIdx.x >> 5;
  int b = blockIdx.x & 31;
  int tid = threadIdx.x, lane = tid & 31, wv = tid >> 5;
  const int N = layerN(l);
  const float* wsrc = (l == 0) ? w0 : ((l == 1) ? w1 : w2);
  const float* bsrc = (l == 0) ? b0 : ((l == 1) ? b1 : b2);
  const float* wrow = wsrc + b * CIN * N;
  const float* brow = bsrc + b * CIN * 64;
  float* wfout = ws + wf_off(l) + b * COUT * N;
  float* bfout = ws + OFF_BF + (l * 32 + b) * COUT * 64;

  __shared__ float sMap[COUT * CIN];
  __shared__ float sBias[COUT];
  for (int i = tid; i < COUT * CIN; i += 128) sMap[i] = mapW[l * COUT * CIN + i];
  if (tid < COUT) sBias[tid] = mapb[l * COUT + tid];
  __syncthreads();

  int mrow = lane & 15, kgrp = lane >> 4;
  int ntiles = N >> 4;
  for (int nt = wv; nt < ntiles; nt += 4) {
    int ncol = nt * 16 + (lane & 15);
    for (int m0 = 0; m0 < 32; m0 += 16) {
      v8f acc;
      for (int j = 0; j < 8; j++) acc[j] = sBias[m0 + j + 8 * kgrp];
      for (int k0 = 0; k0 < 16; k0 += 4) {
        int ka = k0 + 2 * kgrp;
        v2f a, bv;
        a.x = sMap[(m0 + mrow) * CIN + ka];
        a.y = sMap[(m0 + mrow) * CIN + ka + 1];
        bv.x = wrow[ka * N + ncol];
        bv.y = wrow[(ka + 1) * N + ncol];
        acc = wmma_f32_4(a, bv, acc);
      }
      for (int j = 0; j < 8; j++) wfout[(m0 + j + 8 * kgrp) * N + ncol] = acc[j];
    }
  }
  // bias rows: 4 tiles, one per wave
  {
    int ncol = wv * 16 + (lane & 15);
    for (int m0 = 0; m0 < 32; m0 += 16) {
      v8f acc;
      for (int j = 0; j < 8; j++) acc[j] = sBias[m0 + j + 8 * kgrp];
      for (int k0 = 0; k0 < 16; k0 += 4) {
        int ka = k0 + 2 * kgrp;
        v2f a, bv;
        a.x = sMap[(m0 + mrow) * CIN + ka];
        a.y = sMap[(m0 + mrow) * CIN + ka + 1];
        bv.x = brow[ka * 64 + ncol];
        bv.y = brow[(ka + 1) * 64 + ncol];
        acc = wmma_f32_4(a, bv, acc);
      }
      for (int j = 0; j < 8; j++) bfout[(m0 + j + 8 * kgrp) * 64 + ncol] = acc[j];
    }
  }
}

// ---------------------------------------------------------------------------
// K2: per-wf-row features. grid 3072 (l*1024 + b*32 + o), block 256
// wstats stride 24: 0 med,1 mad,2..6 q5,7 mean,8 std,9 skew,10 kurt,11 iqr,
// 12 cv,13..15 specMag,16..18 specPh,19 normw,20 rowsum,21 rowmin,22 rowmax
// ---------------------------------------------------------------------------
__global__ __launch_bounds__(256) void k_roww(float* __restrict__ ws) {
  int gr = blockIdx.x;
  int l = gr >> 10;
  int r = gr & 1023;
  const int N = layerN(l);
  const int logN = (l == 0) ? 7 : 12;
  const float* row = ws + wf_off(l) + r * N;
  float* wst = ws + OFF_WST + gr * 24;
  float* blk = ws + OFF_BLK + gr * 64;
  int tid = threadIdx.x;

  __shared__ float sre[4096];
  __shared__ float sim[4096];
  __shared__ float red[256];
  __shared__ float cs[64];
  __shared__ float sh_mean, sh_med;

  for (int i = tid; i < N; i += 256) sre[i] = row[i];
  __syncthreads();

  // chunk sums for corr pooling (64 chunks of N/64)
  int L = N >> 6;
  if (tid < 64) {
    float s = 0.f;
    for (int k = 0; k < L; k++) s += sre[tid * L + k];
    cs[tid] = s; blk[tid] = s;
  }
  // min / max
  float mn = FLT_BIG, mx = -FLT_BIG;
  for (int i = tid; i < N; i += 256) { float x = sre[i]; mn = fminf(mn, x); mx = fmaxf(mx, x); }
  float rowmin = blockReduceMin(mn, red, tid, 256);
  float rowmax = blockReduceMax(mx, red, tid, 256);
  __syncthreads();
  if (tid == 0) {
    float s = 0.f;
    for (int k = 0; k < 64; k++) s += cs[k];
    sh_mean = s / N;
    wst[20] = s; wst[21] = rowmin; wst[22] = rowmax;
  }
  __syncthreads();
  float mean = sh_mean;

  // centered moments
  float a2 = 0.f, a3 = 0.f, a4 = 0.f;
  for (int i = tid; i < N; i += 256) {
    float d = sre[i] - mean; float d2 = d * d;
    a2 += d2; a3 += d2 * d; a4 += d2 * d2;
  }
  float M2 = blockReduceSum(a2, red, tid, 256);
  float M3 = blockReduceSum(a3, red, tid, 256);
  float M4 = blockReduceSum(a4, red, tid, 256);
  if (tid == 0) {
    float stdv = sqrtf(M2 / (float)(N - 1)) + 1e-8f;
    wst[7] = mean;
    wst[8] = stdv;
    wst[9] = (M3 / N) / (stdv * stdv * stdv);
    wst[10] = (M4 / N) / (stdv * stdv * stdv * stdv);
    wst[12] = stdv / (fabsf(mean) + 1e-8f);
    wst[19] = fmaxf(sqrtf(M2), 1e-12f);
  }
  __syncthreads();

  // sort -> median, quantiles, iqr
  bitonic_sort(sre, N, tid, 256);
  if (tid == 0) {
    sh_med = sre[(N - 1) >> 1];
    wst[0] = sh_med;
    for (int qi = 0; qi < 5; qi++) {
      float pos = 0.25f * qi * (float)(N - 1);
      int i0 = (int)floorf(pos);
      float fr = pos - (float)i0;
      int i1 = (i0 + 1 < N) ? i0 + 1 : N - 1;
      wst[2 + qi] = sre[i0] + fr * (sre[i1] - sre[i0]);
    }
    wst[11] = wst[5] - wst[3];  // q75 - q25
  }
  __syncthreads();
  float med = sh_med;

  // MAD: sort |x - med|
  for (int i = tid; i < N; i += 256) sre[i] = fabsf(row[i] - med);
  __syncthreads();
  bitonic_sort(sre, N, tid, 256);
  if (tid == 0) wst[1] = sre[(N - 1) >> 1];
  __syncthreads();

  // rFFT (full complex radix-2 DIT, imag=0 input)
  for (int i = tid; i < N; i += 256) {
    int rv = (int)(__brev((unsigned)i) >> (32 - logN));
    sre[rv] = row[i];
    sim[rv] = 0.f;
  }
  __syncthreads();
  for (int s = 1; s <= logN; s++) {
    int m = 1 << s, half = m >> 1;
    for (int j = tid; j < (N >> 1); j += 256) {
      int grp = j / half, pos = j - grp * half;
      int i0 = grp * m + pos, i1 = i0 + half;
      float ang = -6.28318530717958647692f * (float)pos / (float)m;
      float wi, wr;
      sincosf(ang, &wi, &wr);
      float tr = wr * sre[i1] - wi * sim[i1];
      float ti = wr * sim[i1] + wi * sre[i1];
      sre[i1] = sre[i0] - tr; sim[i1] = sim[i0] - ti;
      sre[i0] += tr;          sim[i0] += ti;
    }
    __syncthreads();
  }
  if (tid == 0) {
    float bv0 = -FLT_BIG, bv1 = -FLT_BIG, bv2 = -FLT_BIG;
    int bi0 = 0, bi1 = 0, bi2 = 0;
    int nb = (N >> 1) + 1;
    for (int k = 0; k < nb; k++) {
      float mg = sqrtf(sre[k] * sre[k] + sim[k] * sim[k]);
      if (mg > bv0)      { bv2 = bv1; bi2 = bi1; bv1 = bv0; bi1 = bi0; bv0 = mg; bi0 = k; }
      else if (mg > bv1) { bv2 = bv1; bi2 = bi1; bv1 = mg; bi1 = k; }
      else if (mg > bv2) { bv2 = mg; bi2 = k; }
    }
    wst[13] = bv0; wst[14] = bv1; wst[15] = bv2;
    wst[16] = atan2f(sim[bi0], sre[bi0]);
    wst[17] = atan2f(sim[bi1], sre[bi1]);
    wst[18] = atan2f(sim[bi2], sre[bi2]);
  }
}

// ---------------------------------------------------------------------------
// K3: per-bf-row features + corr. grid 3072, block 64
// bstats stride 18: 0 med,1 mad,2..6 q5,7..9 specMag,10..12 specPh,
// 13 skew,14 kurt,15 iqr,16 cv,17 corr
// ---------------------------------------------------------------------------
__global__ __launch_bounds__(64) void k_rowb(float* __restrict__ ws) {
  int gr = blockIdx.x;
  int l = gr >> 10;
  const float* row = ws + OFF_BF + gr * 64;
  float* bst = ws + OFF_BST + gr * 18;
  const float* wst = ws + OFF_WST + gr * 24;
  const float* blk = ws + OFF_BLK + gr * 64;
  int tid = threadIdx.x;
  const int N = 64, logN = 6;

  __shared__ float sre[64], sim[64], red[64];
  __shared__ float sh_mean, sh_med;

  float x = row[tid];
  sre[tid] = x;
  float sum = blockReduceSum(x, red, tid, 64);
  if (tid == 0) sh_mean = sum / 64.f;
  __syncthreads();
  float mean = sh_mean;
  float d = x - mean, d2 = d * d;
  float M2 = blockReduceSum(d2, red, tid, 64);
  float M3 = blockReduceSum(d2 * d, red, tid, 64);
  float M4 = blockReduceSum(d2 * d2, red, tid, 64);
  if (tid == 0) {
    float stdv = sqrtf(M2 / 63.f) + 1e-8f;
    bst[13] = (M3 / 64.f) / (stdv * stdv * stdv);
    bst[14] = (M4 / 64.f) / (stdv * stdv * stdv * stdv);
    bst[16] = stdv / (fabsf(mean) + 1e-8f);
    // correlation against pooled-normalized wf row
    float normb = fmaxf(sqrtf(M2), 1e-12f);
    float meanw = wst[7];
    float normw = wst[19];
    float Lw = (l == 0) ? 2.f : 64.f;
    float c = 0.f;
    for (int t = 0; t < 64; t++) {
      float pw = (blk[t] - Lw * meanw) / (Lw * normw);
      float pb = (row[t] - mean) / normb;
      c += pw * pb;
    }
    bst[17] = c;
  }
  __syncthreads();

  bitonic_sort(sre, N, tid, 64);
  if (tid == 0) {
    sh_med = sre[31];
    bst[0] = sh_med;
    for (int qi = 0; qi < 5; qi++) {
      float pos = 0.25f * qi * 63.f;
      int i0 = (int)floorf(pos);
      float fr = pos - (float)i0;
      int i1 = (i0 + 1 < N) ? i0 + 1 : N - 1;
      bst[2 + qi] = sre[i0] + fr * (sre[i1] - sre[i0]);
    }
    bst[15] = bst[5] - bst[3];
  }
  __syncthreads();
  sre[tid] = fabsf(row[tid] - sh_med);
  __syncthreads();
  bitonic_sort(sre, N, tid, 64);
  if (tid == 0) bst[1] = sre[31];
  __syncthreads();

  // FFT(64)
  int rv = (int)(__brev((unsigned)tid) >> (32 - logN));
  sre[rv] = row[tid];
  sim[rv] = 0.f;
  __syncthreads();
  for (int s = 1; s <= logN; s++) {
    int m = 1 << s, half = m >> 1;
    for (int j = tid; j < (N >> 1); j += 64) {
      int grp = j / half, pos = j - grp * half;
      int i0 = grp * m + pos, i1 = i0 + half;
      float ang = -6.28318530717958647692f * (float)pos / (float)m;
      float wi, wr;
      sincosf(ang, &wi, &wr);
      float tr = wr * sre[i1] - wi * sim[i1];
      float ti = wr * sim[i1] + wi * sre[i1];
      sre[i1] = sre[i0] - tr; sim[i1] = sim[i0] - ti;
      sre[i0] += tr;          sim[i0] += ti;
    }
    __syncthreads();
  }
  if (tid == 0) {
    float bv0 = -FLT_BIG, bv1 = -FLT_BIG, bv2 = -FLT_BIG;
    int bi0 = 0, bi1 = 0, bi2 = 0;
    for (int k = 0; k < 33; k++) {
      float mg = sqrtf(sre[k] * sre[k] + sim[k] * sim[k]);
      if (mg > bv0)      { bv2 = bv1; bi2 = bi1; bv1 = bv0; bi1 = bi0; bv0 = mg; bi0 = k; }
      else if (mg > bv1) { bv2 = bv1; bi2 = bi1; bv1 = mg; bi1 = k; }
      else if (mg > bv2) { bv2 = mg; bi2 = k; }
    }
    bst[7] = bv0; bst[8] = bv1; bst[9] = bv2;
    bst[10] = atan2f(sim[bi0], sre[bi0]);
    bst[11] = atan2f(sim[bi1], sre[bi1]);
    bst[12] = atan2f(sim[bi2], sre[bi2]);
  }
}

// ---------------------------------------------------------------------------
// K4: global stats per (l,b): gmean/gmin/gmax reduce + exact global median/MAD
// via 32-step bisection on monotone float keys. grid 96, block 256
// ---------------------------------------------------------------------------
__global__ __launch_bounds__(256) void k_gmad(float* __restrict__ ws) {
  int l = blockIdx.x >> 5, b = blockIdx.x & 31;
  const int N = layerN(l);
  const int n = 32 * N;
  const float* base = ws + wf_off(l) + b * 32 * N;
  float* g = ws + OFF_GST + (l * 32 + b) * 4;
  int tid = threadIdx.x;
  __shared__ int redi[256];

  if (tid == 0) {
    const float* wst0 = ws + OFF_WST + (l * 1024 + b * 32) * 24;
    float s = 0.f, mn = FLT_BIG, mx = -FLT_BIG;
    for (int o = 0; o < 32; o++) {
      s += wst0[o * 24 + 20];
      mn = fminf(mn, wst0[o * 24 + 21]);
      mx = fmaxf(mx, wst0[o * 24 + 22]);
    }
    g[0] = s / (float)n; g[2] = mn; g[3] = mx;
  }
  int target = ((n - 1) >> 1) + 1;

  unsigned lo = 0u, hi = 0xFFFFFFFFu;
  while (lo < hi) {
    unsigned mid = lo + ((hi - lo) >> 1);
    int c = 0;
    for (int i = tid; i < n; i += 256) c += (fkey(base[i]) <= mid) ? 1 : 0;
    int tot = blockReduceSumI(c, redi, tid, 256);
    __syncthreads();
    if (tot >= target) hi = mid; else lo = mid + 1;
  }
  float med = funkey(lo);

  lo = 0u; hi = 0xFFFFFFFFu;
  while (lo < hi) {
    unsigned mid = lo + ((hi - lo) >> 1);
    int c = 0;
    for (int i = tid; i < n; i += 256) c += (fkey(fabsf(base[i] - med)) <= mid) ? 1 : 0;
    int tot = blockReduceSumI(c, redi, tid, 256);
    __syncthreads();
    if (tot >= target) hi = mid; else lo = mid + 1;
  }
  if (tid == 0) g[1] = funkey(lo);
}

// ---------------------------------------------------------------------------
// feature gather (order matches reference concatenation)
// ---------------------------------------------------------------------------
__device__ __forceinline__ float get_feat(const float* ws, int l, int r, int f) {
  int gr = l * 1024 + r;
  const float* wst = ws + OFF_WST + gr * 24;
  const float* bst = ws + OFF_BST + gr * 18;
  const float* g = ws + OFF_GST + (l * 32 + (r >> 5)) * 4;
  if (f == 0) return wst[0];
  if (f == 1) return wst[1];
  if (f < 7)  return wst[2 + (f - 2)];
  if (f == 7) return bst[0];
  if (f == 8) return bst[1];
  if (f < 14) return bst[2 + (f - 9)];
  if (f < 18) return g[f - 14];
  if (f < 24) return wst[13 + (f - 18)];
  if (f < 30) return bst[7 + (f - 24)];
  if (f == 30) return bst[17];
  if (f == 31) return wst[9];
  if (f == 32) return wst[10];
  if (f == 33) return wst[11];
  if (f == 34) return wst[12];
  return bst[13 + (f - 35)];
}

// ---------------------------------------------------------------------------
// K5: LN(39) + ResidualMLP with f32 WMMA. grid 3*64 tiles of 16 rows, block 128
// ---------------------------------------------------------------------------
__global__ __launch_bounds__(128) void k_mlp(
    const float* __restrict__ ws,
    const float* __restrict__ fc1W, const float* __restrict__ fc1b,
    const float* __restrict__ lng, const float* __restrict__ lnb,
    const float* __restrict__ gateW, const float* __restrict__ gateb,
    const float* __restrict__ outW, const float* __restrict__ outb,
    const float* __restrict__ scale, float* __restrict__ out) {
  int l = blockIdx.x / 64;
  int tile = blockIdx.x % 64;
  int r0 = tile * 16;
  int tid = threadIdx.x, lane = tid & 31, wv = tid >> 5;
  int mrow = lane & 15, kgrp = lane >> 4;

  __shared__ float sA[16][40];
  __shared__ float sH[16][64];
  __shared__ float sG[16][64];
  __shared__ float sMean[16], sRstd[16];

  // gather + LN over 39 features
  for (int idx = tid; idx < 16 * F_; idx += 128) {
    int rr = idx / F_, f = idx % F_;
    sA[rr][f] = get_feat(ws, l, r0 + rr, f);
  }
  __syncthreads();
  if (tid < 16) {
    float m = 0.f;
    for (int f = 0; f < F_; f++) m += sA[tid][f];
    m /= (float)F_;
    float v = 0.f;
    for (int f = 0; f < F_; f++) { float dd = sA[tid][f] - m; v += dd * dd; }
    v /= (float)F_;
    sMean[tid] = m; sRstd[tid] = rsqrtf(v + 1e-5f);
  }
  __syncthreads();
  for (int idx = tid; idx < 16 * 40; idx += 128) {
    int rr = idx / 40, f = idx % 40;
    float x = (f < F_) ? (sA[rr][f] - sMean[rr]) * sRstd[rr] : 0.f;
    sA[rr][f] = x;
  }
  __syncthreads();

  // fc1: h[16x64] = sA[16x40] @ fc1W^T[40x64] + fc1b ; K padded to 40
  {
    int n0 = wv * 16;
    int ncol = n0 + (lane & 15);
    v8f acc;
    float bias = fc1b[l * H_ + ncol];
    for (int j = 0; j < 8; j++) acc[j] = bias;
    for (int k0 = 0; k0 < 40; k0 += 4) {
      int ka = k0 + 2 * kgrp;
      v2f a, bv;
      a.x = sA[mrow][ka]; a.y = sA[mrow][ka + 1];
      bv.x = (ka < F_)     ? fc1W[(l * H_ + ncol) * F_ + ka]     : 0.f;
      bv.y = (ka + 1 < F_) ? fc1W[(l * H_ + ncol) * F_ + ka + 1] : 0.f;
      acc = wmma_f32_4(a, bv, acc);
    }
    for (int j = 0; j < 8; j++) sH[j + 8 * kgrp][ncol] = acc[j];
  }
  __syncthreads();

  // LN * lng + lnb, exact gelu
  if (tid < 16) {
    float m = 0.f;
    for (int c = 0; c < H_; c++) m += sH[tid][c];
    m /= (float)H_;
    float v = 0.f;
    for (int c = 0; c < H_; c++) { float dd = sH[tid][c] - m; v += dd * dd; }
    v /= (float)H_;
    sMean[tid] = m; sRstd[tid] = rsqrtf(v + 1e-5f);
  }
  __syncthreads();
  for (int idx = tid; idx < 16 * H_; idx += 128) {
    int rr = idx >> 6, c = idx & 63;
    float x = (sH[rr][c] - sMean[rr]) * sRstd[rr] * lng[l * H_ + c] + lnb[l * H_ + c];
    x = 0.5f * x * (1.0f + erff(x * 0.70710678118654752440f));
    sH[rr][c] = x;
  }
  __syncthreads();

  // gate logits: z[16x64] = sH @ gateW^T + gateb
  {
    int n0 = wv * 16;
    int ncol = n0 + (lane & 15);
    v8f acc;
    float bias = gateb[l * H_ + ncol];
    for (int j = 0; j < 8; j++) acc[j] = bias;
    for (int k0 = 0; k0 < H_; k0 += 4) {
      int ka = k0 + 2 * kgrp;
      v2f a, bv;
      a.x = sH[mrow][ka]; a.y = sH[mrow][ka + 1];
      bv.x = gateW[(l * H_ + ncol) * H_ + ka];
      bv.y = gateW[(l * H_ + ncol) * H_ + ka + 1];
      acc = wmma_f32_4(a, bv, acc);
    }
    for (int j = 0; j < 8; j++) sG[j + 8 * kgrp][ncol] = acc[j];
  }
  __syncthreads();
  for (int idx = tid; idx < 16 * H_; idx += 128) {
    int rr = idx >> 6, c = idx & 63;
    float z = sG[rr][c];
    sG[rr][c] = sH[rr][c] / (1.f + expf(-z));
  }
  __syncthreads();

  // out: raw[16x256] = sG @ outW^T + outb ; then sin*scale + 1
  float scl = scale[l];
  for (int t = 0; t < 4; t++) {
    int n0 = (wv + 4 * t) * 16;
    int ncol = n0 + (lane & 15);
    v8f acc;
    float bias = outb[l * D_ + ncol];
    for (int j = 0; j < 8; j++) acc[j] = bias;
    for (int k0 = 0; k0 < H_; k0 += 4) {
      int ka = k0 + 2 * kgrp;
      v2f a, bv;
      a.x = sG[mrow][ka]; a.y = sG[mrow][ka + 1];
      bv.x = outW[(l * D_ + ncol) * H_ + ka];
      bv.y = outW[(l * D_ + ncol) * H_ + ka + 1];
      acc = wmma_f32_4(a, bv, acc);
    }
    for (int j = 0; j < 8; j++) {
      int row = r0 + j + 8 * kgrp;
      out[(l * 1024 + row) * D_ + ncol] = sinf(acc[j]) * scl + 1.0f;
    }
  }
}

// ---------------------------------------------------------------------------
// launch
// ---------------------------------------------------------------------------
extern "C" void kernel_launch(void* const* d_in, const int* in_sizes, int n_in,
                              void* d_out, int out_size, void* d_ws, size_t ws_size,
                              hipStream_t stream) {
  const float* w0    = (const float*)d_in[0];
  const float* b0    = (const float*)d_in[1];
  const float* w1    = (const float*)d_in[2];
  const float* b1    = (const float*)d_in[3];
  const float* w2    = (const float*)d_in[4];
  const float* b2    = (const float*)d_in[5];
  const float* mapW  = (const float*)d_in[6];
  const float* mapb  = (const float*)d_in[7];
  const float* fc1W  = (const float*)d_in[8];
  const float* fc1b  = (const float*)d_in[9];
  const float* lng   = (const float*)d_in[10];
  const float* lnb   = (const float*)d_in[11];
  const float* gateW = (const float*)d_in[12];
  const float* gateb = (const float*)d_in[13];
  const float* outW  = (const float*)d_in[14];
  const float* outb  = (const float*)d_in[15];
  const float* scl   = (const float*)d_in[16];
  float* ws = (float*)d_ws;
  float* out = (float*)d_out;

  k_proj<<<96, 128, 0, stream>>>(w0, b0, w1, b1, w2, b2, mapW, mapb, ws);
  k_roww<<<3072, 256, 0, stream>>>(ws);
  k_rowb<<<3072, 64, 0, stream>>>(ws);
  k_gmad<<<96, 256, 0, stream>>>(ws);
  k_mlp<<<192, 128, 0, stream>>>(ws, fc1W, fc1b, lng, lnb, gateW, gateb,
                                 outW, outb, scl, out);
}